// EarthAttention3D_45586782879921
// MI455X (gfx1250) — compile-verified
//
#include <hip/hip_runtime.h>
#include <cstdint>
#include <cstddef>

typedef float v2f __attribute__((ext_vector_type(2)));
typedef float v8f __attribute__((ext_vector_type(8)));
typedef unsigned int u32x4 __attribute__((ext_vector_type(4)));
typedef int i32x8 __attribute__((ext_vector_type(8)));
typedef int i32x4 __attribute__((ext_vector_type(4)));

// D(16x16,f32) = A(16x4,f32) x B(4x16,f32) + C   (CDNA5 V_WMMA_F32_16X16X4_F32)
__device__ __forceinline__ v8f wmma_f32_k4(v2f a, v2f b, v8f c) {
  return __builtin_amdgcn_wmma_f32_16x16x4_f32(
      /*neg_a=*/false, a, /*neg_b=*/false, b,
      /*c_mod=*/(short)0, c, /*reuse_a=*/false, /*reuse_b=*/false);
}

constexpr int  kB   = 15;    // B_ (== nLon, B == 1)
constexpr int  kNW  = 64;    // windows
constexpr int  kN   = 144;   // tokens per window
constexpr int  kC   = 192;   // channels
constexpr int  kH   = 6;     // heads
constexpr int  kL   = 32;    // head dim
constexpr int  kRows = kB * kNW * kN;               // 138240
constexpr long long kTSZ = (long long)kB * kH * kNW * kN * kL; // elems per q/k/v plane
constexpr int  kSStr = 146;                         // padded LDS stride for S (even -> b64 ok)
constexpr int  kKVStr = 34;                         // TDM-padded LDS stride for K/V tiles
constexpr int  kKVFloats = kN * kKVStr;             // 4896 floats per K/V LDS image
constexpr int  kSmemAttn = (2 * kKVFloats + kN * kSStr) * 4; // 123264 B dynamic LDS

#if defined(__HIP_DEVICE_COMPILE__) && __has_builtin(__builtin_amdgcn_tensor_load_to_lds) && __has_builtin(__builtin_amdgcn_s_wait_tensorcnt)
#define USE_TDM 1
#else
#define USE_TDM 0
#endif

#if USE_TDM
// Issue one TDM descriptor: copy kN*kL f32 (row-major, 32 f32/row) from
// `gaddr` into LDS byte offset `ldsOff`, inserting 2 DWORDs of pad every
// 32 DWORDs (-> LDS row stride 34 floats, conflict-free b64 reads).
__device__ __forceinline__ void tdm_load_tile(uint64_t gaddr, uint32_t ldsOff) {
  constexpr unsigned elems = (unsigned)(kN * kL);   // 4608, 1-D tile
  u32x4 g0;
  g0[0] = 1u;                                       // count=1 (valid), user mode
  g0[1] = ldsOff;                                   // lds_addr (bytes)
  g0[2] = (uint32_t)gaddr;                          // global_addr[31:0]
  g0[3] = (uint32_t)((gaddr >> 32) & 0x01FFFFFFu) | (2u << 30); // addr[56:32] | type=2
  i32x8 g1;
  g1[0] = (int)((2u << 16)        // data_size = 4B
              | (1u << 20)        // pad_enable
              | (4u << 22)        // pad_interval code 4 -> every 32 DWORDs
              | (1u << 25));      // pad_amount  code 1 -> 2 DWORDs
  g1[1] = (int)((elems & 0xFFFFu) << 16);           // tensor_dim0[15:0] @ bits 63:48
  g1[2] = (int)((elems >> 16) | (1u << 16));        // tensor_dim0[31:16] | tensor_dim1=1
  g1[3] = (int)(elems << 16);                       // tile_dim0 @ bits 127:112
  g1[4] = 0;                                        // tile_dim1=0 (1-D), tile_dim2=0
  g1[5] = (int)elems;                               // tensor_dim0_stride (unused for 1-D)
  g1[6] = 0;
  g1[7] = 0;
  i32x4 z4 = {};
  i32x8 z8 = {};
  // 6-arg variant (clang-23 / therock-10.0 headers)
  __builtin_amdgcn_tensor_load_to_lds(g0, g1, z4, z4, z8, 0);
}
#endif

// ---------------------------------------------------------------------------
// Kernel 1: QKV = x @ qkv_w^T + qkv_b, scattered to (t, b_, h, w, n, l)
// grid = 8640 (one 16-row M tile), block = 288 (9 waves x 64 output cols)
// ---------------------------------------------------------------------------
__global__ __launch_bounds__(288) void qkv_kernel(
    const float* __restrict__ x, const float* __restrict__ wmat,
    const float* __restrict__ bvec, float* __restrict__ qkv) {
  __shared__ float xs[16 * 194];
  const int tid = threadIdx.x;
  const int m0  = blockIdx.x * 16;

  for (int e = tid; e < 16 * kC; e += 288) {
    const int m = e / kC, c = e - m * kC;
    xs[m * 194 + c] = x[(size_t)(m0 + m) * kC + c];
  }
  __syncthreads();

  const int wave = tid >> 5, lane = tid & 31;
  const int lh = lane & 15, half = lane >> 4;
  const int obase = wave * 64;

  v8f acc[4] = {};
  for (int k = 0; k < kC; k += 4) {
    const v2f a = *(const v2f*)&xs[lh * 194 + k + half * 2];
#pragma unroll
    for (int t = 0; t < 4; ++t) {
      const int o = obase + t * 16 + lh;
      const v2f b = *(const v2f*)&wmat[(size_t)o * kC + k + half * 2];
      acc[t] = wmma_f32_k4(a, b, acc[t]);
    }
  }

  const int bw  = m0 / kN;             // (b_, w) window id (m-tiles never straddle)
  const int b_  = bw / kNW, wid = bw - b_ * kNW;
  const int n0  = m0 - bw * kN;
#pragma unroll
  for (int t = 0; t < 4; ++t) {
    const int o   = obase + t * 16 + lh;
    const int tq  = o / kC;
    const int rem = o - tq * kC;
    const int hh  = rem >> 5, l = rem & 31;
    const float bb = bvec[o];
    float* dst = qkv +
        ((((size_t)tq * kB + b_) * kH + hh) * kNW + wid) * ((size_t)kN * kL) + l;
#pragma unroll
    for (int r = 0; r < 8; ++r) {
      const int n = n0 + r + half * 8;
      dst[(size_t)n * kL] = acc[t][r] + bb;
    }
  }
}

// ---------------------------------------------------------------------------
// Kernel 2: fused attention per (b_, h, w): S = scale*QK^T + bias + mask,
// softmax(S) in LDS, O = S @ V written over the Q region.
// K and V tiles staged to LDS once per workgroup via the Tensor Data Mover
// (TENSORcnt), with D# padding giving a conflict-free 34-float LDS stride.
// grid = 5760, block = 288 (9 waves, one 16-row strip each)
// ---------------------------------------------------------------------------
__global__ __launch_bounds__(288) void attn_kernel(
    float* __restrict__ qkv, const float* __restrict__ mask,
    const float* __restrict__ btab) {
  extern __shared__ float lds[];
  float* ks = lds;                       // kN x 34 (TDM padded)
  float* vs = lds + kKVFloats;           // kN x 34 (TDM padded)
  float* Ss = lds + 2 * kKVFloats;       // kN x 146

  const int tid = threadIdx.x;
  const int wave = tid >> 5, lane = tid & 31;
  const int lh = lane & 15, half = lane >> 4;

  const int blk = blockIdx.x;
  const int b_  = blk / (kH * kNW);
  const int rem = blk - b_ * kH * kNW;
  const int h   = rem / kNW;
  const int w   = rem - h * kNW;

  float* qp       = qkv + (((size_t)b_ * kH + h) * kNW + w) * ((size_t)kN * kL);
  const float* kp = qp + kTSZ;
  const float* vp = qp + 2 * kTSZ;

#if USE_TDM
  if (wave == 0) {                       // one wave issues both TDM descriptors
    tdm_load_tile((uint64_t)(uintptr_t)kp, 0u);
    tdm_load_tile((uint64_t)(uintptr_t)vp, (uint32_t)(kKVFloats * 4));
    __builtin_amdgcn_s_wait_tensorcnt(0);
  }
#else
  for (int e = tid; e < kN * kL; e += 288) {
    const int r = e >> 5, c = e & 31;
    ks[r * kKVStr + c] = kp[e];
    vs[r * kKVStr + c] = vp[e];
  }
#endif
  __syncthreads();

  const int m0 = wave * 16;

  // ---- S = Q K^T (Q frags from global, K frags from LDS) -------------------
  v8f acc[9] = {};
  for (int k = 0; k < kL; k += 4) {
    const v2f a = *(const v2f*)&qp[(m0 + lh) * kL + k + half * 2];
#pragma unroll
    for (int t = 0; t < 9; ++t) {
      const v2f b = *(const v2f*)&ks[(t * 16 + lh) * kKVStr + k + half * 2];
      acc[t] = wmma_f32_k4(a, b, acc[t]);
    }
  }

  const float scale = 0.17677669529663689f; // 32^-0.5
  const float* mrow = mask + (((size_t)(b_ % kB) * kNW + w) * kN) * kN;
  const int wh = w * kH + h;

#pragma unroll
  for (int t = 0; t < 9; ++t) {
    const int col = t * 16 + lh;
    const int jz = col / 72, jh = (col / 12) % 6, jw = col % 12;
    const int cterm = 828 * 2 * jz + 23 * 6 * jh - jw + 11;
#pragma unroll
    for (int r = 0; r < 8; ++r) {
      const int row = m0 + r + half * 8;
      const int iz = row / 72, ih = (row / 12) % 6, iw = row % 12;
      const int epi = 828 * iz + 23 * ih + iw + cterm;
      const float bias = btab[(size_t)epi * (kNW * kH) + wh];
      const float mk   = mrow[(size_t)row * kN + col];
      Ss[row * kSStr + col] = acc[t][r] * scale + bias + mk;
    }
  }
  __syncthreads();

  // ---- row softmax ---------------------------------------------------------
  if (tid < kN) {
    float* srow = Ss + tid * kSStr;
    float mx = srow[0];
    for (int j = 1; j < kN; ++j) mx = fmaxf(mx, srow[j]);
    float sum = 0.f;
    for (int j = 0; j < kN; ++j) { const float e = __expf(srow[j] - mx); srow[j] = e; sum += e; }
    const float inv = 1.f / sum;
    for (int j = 0; j < kN; ++j) srow[j] *= inv;
  }
  __syncthreads();

  // ---- O = P V (V frags from LDS), write over Q ----------------------------
  v8f oacc[2] = {};
  for (int k = 0; k < kN; k += 4) {
    const v2f a = *(const v2f*)&Ss[(m0 + lh) * kSStr + k + half * 2];
#pragma unroll
    for (int t = 0; t < 2; ++t) {
      v2f b;
      b.x = vs[(k + half * 2 + 0) * kKVStr + t * 16 + lh];
      b.y = vs[(k + half * 2 + 1) * kKVStr + t * 16 + lh];
      oacc[t] = wmma_f32_k4(a, b, oacc[t]);
    }
  }
#pragma unroll
  for (int t = 0; t < 2; ++t) {
    const int l = t * 16 + lh;
#pragma unroll
    for (int r = 0; r < 8; ++r) {
      const int n = m0 + r + half * 8;
      qp[n * kL + l] = oacc[t][r];
    }
  }
}

// ---------------------------------------------------------------------------
// Kernel 3: out = gather(O) @ proj_w^T + proj_b
// grid = 8640, block = 192 (6 waves x 32 output cols)
// ---------------------------------------------------------------------------
__global__ __launch_bounds__(192) void proj_kernel(
    const float* __restrict__ ows, const float* __restrict__ wmat,
    const float* __restrict__ bvec, float* __restrict__ out) {
  __shared__ float xs[16 * 194];
  const int tid = threadIdx.x;
  const int m0  = blockIdx.x * 16;
  const int bw  = m0 / kN;
  const int b_  = bw / kNW, wid = bw - b_ * kNW;
  const int n0  = m0 - bw * kN;

  for (int e = tid; e < 16 * kC; e += 192) {
    const int m = e / kC, c = e - m * kC;
    const int hh = c >> 5, l = c & 31;
    xs[m * 194 + c] =
        ows[(((size_t)b_ * kH + hh) * kNW + wid) * ((size_t)kN * kL) +
            (size_t)(n0 + m) * kL + l];
  }
  __syncthreads();

  const int wave = tid >> 5, lane = tid & 31;
  const int lh = lane & 15, half = lane >> 4;
  const int obase = wave * 32;

  v8f acc[2] = {};
  for (int k = 0; k < kC; k += 4) {
    const v2f a = *(const v2f*)&xs[lh * 194 + k + half * 2];
#pragma unroll
    for (int t = 0; t < 2; ++t) {
      const int o = obase + t * 16 + lh;
      const v2f b = *(const v2f*)&wmat[(size_t)o * kC + k + half * 2];
      acc[t] = wmma_f32_k4(a, b, acc[t]);
    }
  }
#pragma unroll
  for (int t = 0; t < 2; ++t) {
    const int o = obase + t * 16 + lh;
    const float bb = bvec[o];
#pragma unroll
    for (int r = 0; r < 8; ++r) {
      const int rowg = m0 + r + half * 8;
      out[(size_t)rowg * kC + o] = acc[t][r] + bb;
    }
  }
}

// ---------------------------------------------------------------------------
extern "C" void kernel_launch(void* const* d_in, const int* in_sizes, int n_in,
                              void* d_out, int out_size, void* d_ws, size_t ws_size,
                              hipStream_t stream) {
  const float* x      = (const float*)d_in[0];
  const float* mask   = (const float*)d_in[1];
  const float* qkv_w  = (const float*)d_in[2];
  const float* qkv_b  = (const float*)d_in[3];
  const float* proj_w = (const float*)d_in[4];
  const float* proj_b = (const float*)d_in[5];
  const float* btab   = (const float*)d_in[6];
  float* out = (float*)d_out;
  float* qkv = (float*)d_ws;   // 3 * kTSZ floats (q | k | v); O overwrites q plane

  // opt-in to 120 KB dynamic LDS (not a stream op; capture-safe, idempotent)
  (void)hipFuncSetAttribute((const void*)attn_kernel,
                            hipFuncAttributeMaxDynamicSharedMemorySize, kSmemAttn);

  qkv_kernel <<<kRows / 16, 288, 0, stream>>>(x, qkv_w, qkv_b, qkv);
  attn_kernel<<<kB * kH * kNW, 288, kSmemAttn, stream>>>(qkv, mask, btab);
  proj_kernel<<<kRows / 16, 192, 0, stream>>>(qkv, proj_w, proj_b, out);
}